// Multi_Scale_Triplane_Encoder_39986145526173
// MI455X (gfx1250) — compile-verified
//
#include <hip/hip_runtime.h>
#include <cstddef>
#include <cstdint>

// ---------------------------------------------------------------------------
// Multi-scale triplane encoder (scatter-mean) for MI455X / gfx1250, wave32.
//
//   B = 8, C = 32, scales: (N=2^16,res=64), (N=2^15,res=128), (N=2^14,res=256)
//   planes per scale in order: xz (axes 0,2), xy (axes 0,1), yz (axes 1,2)
//   plane output element: out[b][c][idx]  with idx = i_a + res * i_b
//   flat output = concat(scale0{xz,xy,yz}, scale1{...}, scale2{...})
//
// wave32 mapping: one wave per point, lane == channel (C==32 exactly).
//  - coords fetched through the SCALAR path (readfirstlane -> s_load),
//    keeping the vector memory pipe free for the 6 atomics per wave.
//  - cell math uses true IEEE f32 division (faithful to the reference; a
//    reciprocal-multiply can flip trunc cells, which is a discrete error).
//  - normalize is 128-bit vectorized with a non-temporal final store so the
//    hot per-batch count rows stay L2-resident.
// ---------------------------------------------------------------------------

#define TE_B 8
#define TE_C 32
#define TE_DENOM (1.0f + 0.1f + 1e-5f)   // 1 + PADDING + EPS
#define TE_CLIPHI (1.0f - 1e-5f)         // 1 - EPS

// Native clang vectors (HIP_vector_type is rejected by nontemporal builtins).
typedef float        te_f4 __attribute__((ext_vector_type(4)));
typedef unsigned int te_u4 __attribute__((ext_vector_type(4)));

template <int RES>
__device__ __forceinline__ int te_cell(float v) {
    float u = v / TE_DENOM + 0.5f;       // IEEE div: bit-faithful cell indices
    u = fminf(fmaxf(u, 0.0f), TE_CLIPHI);
    return (int)(u * (float)RES);        // trunc; u >= 0 so == floor
}

// One wave (32 lanes) per point. Lane = channel.
// outScale -> base of this scale's 3 plane outputs (each B*C*R2 floats)
// cntScale -> base of this scale's 3 count planes  (each B*R2 uints)
template <int NLOG2, int RES>
__global__ __launch_bounds__(256)
void te_scatter_kernel(const float* __restrict__ feats,   // (B,N,32)
                       const float* __restrict__ coords,  // (B,N,3)
                       float* __restrict__ outScale,
                       unsigned int* __restrict__ cntScale) {
    constexpr int R2 = RES * RES;
    constexpr size_t planeOut = (size_t)TE_B * TE_C * (size_t)R2;
    constexpr size_t planeCnt = (size_t)TE_B * (size_t)R2;

    const int gid  = blockIdx.x * blockDim.x + threadIdx.x;
    const int lane = gid & 31;                         // channel
    // Point id is identical for all 32 lanes of the wave: pin it to an SGPR
    // so the coord fetch can go down the scalar (constant-cache) path.
    const int pt   = __builtin_amdgcn_readfirstlane(gid >> 5);
    const int b    = pt >> NLOG2;                      // batch (shift, no idiv)

    const float* cp = coords + (size_t)pt * 3;         // uniform address
    const float x = cp[0];
    const float y = cp[1];
    const float z = cp[2];

    // Coalesced: 32 lanes read 32 consecutive floats (128B per wave).
    const float f = feats[((size_t)pt << 5) + lane];

    const int ix = te_cell<RES>(x);
    const int iy = te_cell<RES>(y);
    const int iz = te_cell<RES>(z);

    const int idx_xz = ix + RES * iz;
    const int idx_xy = ix + RES * iy;
    const int idx_yz = iy + RES * iz;

    const size_t rowBase = (size_t)(b * TE_C + lane) * (size_t)R2;

    // Hardware f32 atomics (no-return, STOREcnt) resolved in L2.
    unsafeAtomicAdd(outScale + 0 * planeOut + rowBase + (size_t)idx_xz, f);
    unsafeAtomicAdd(outScale + 1 * planeOut + rowBase + (size_t)idx_xy, f);
    unsafeAtomicAdd(outScale + 2 * planeOut + rowBase + (size_t)idx_yz, f);

    if (lane == 0) {
        const size_t cb = (size_t)b * (size_t)R2;
        atomicAdd(cntScale + 0 * planeCnt + cb + (size_t)idx_xz, 1u);
        atomicAdd(cntScale + 1 * planeCnt + cb + (size_t)idx_xy, 1u);
        atomicAdd(cntScale + 2 * planeCnt + cb + (size_t)idx_yz, 1u);
    }
}

// Elementwise: plane[e] /= max(count[b, idx], 1).
// 4 elements per thread -> global_load_b128 / global_store_b128.
// Final store is non-temporal: the normalized plane is never re-read on
// device, while the count rows are re-read 32x and should own L2/WGP$.
template <int RES>
__global__ __launch_bounds__(256)
void te_normalize_kernel(float* __restrict__ plane,
                         const unsigned int* __restrict__ cnt) {
    constexpr int R2 = RES * RES;
    constexpr int R2LOG2  = __builtin_ctz(R2);
    constexpr int ROWLOG2 = 5 + R2LOG2;               // log2(C*R2)

    const size_t e4   = (size_t)blockIdx.x * blockDim.x + threadIdx.x;
    const size_t base = e4 << 2;                      // grid sized exactly
    const int b   = (int)(base >> ROWLOG2);
    const int idx = (int)(base & (size_t)(R2 - 1));

    const te_u4 c4 = *(const te_u4*)(cnt + ((size_t)b << R2LOG2) + idx);
    te_f4 v = *(const te_f4*)(plane + base);

    v.x = v.x / (float)(c4.x > 1u ? c4.x : 1u);
    v.y = v.y / (float)(c4.y > 1u ? c4.y : 1u);
    v.z = v.z / (float)(c4.z > 1u ? c4.z : 1u);
    v.w = v.w / (float)(c4.w > 1u ? c4.w : 1u);

    __builtin_nontemporal_store(v, (te_f4*)(plane + base));
}

extern "C" void kernel_launch(void* const* d_in, const int* in_sizes, int n_in,
                              void* d_out, int out_size, void* d_ws, size_t ws_size,
                              hipStream_t stream) {
    // setup_inputs() dict order: f0, c0, f1, c1, f2, c2
    const float* F[3]  = {(const float*)d_in[0], (const float*)d_in[2], (const float*)d_in[4]};
    const float* Cc[3] = {(const float*)d_in[1], (const float*)d_in[3], (const float*)d_in[5]};

    float* out = (float*)d_out;
    unsigned int* cnt = (unsigned int*)d_ws;

    // Per-plane sizes.
    const size_t P[3] = {(size_t)TE_B * TE_C * 64 * 64,
                         (size_t)TE_B * TE_C * 128 * 128,
                         (size_t)TE_B * TE_C * 256 * 256};
    const size_t Q[3] = {(size_t)TE_B * 64 * 64,
                         (size_t)TE_B * 128 * 128,
                         (size_t)TE_B * 256 * 256};

    float* outS[3];
    unsigned int* cntS[3];
    outS[0] = out;
    outS[1] = out + 3 * P[0];
    outS[2] = out + 3 * P[0] + 3 * P[1];
    cntS[0] = cnt;
    cntS[1] = cnt + 3 * Q[0];
    cntS[2] = cnt + 3 * Q[0] + 3 * Q[1];

    const size_t cntElems = 3 * (Q[0] + Q[1] + Q[2]);   // 2,064,384 u32 (~8.25 MB)

    // Zero accumulators (graph-capture-safe memset nodes).
    (void)hipMemsetAsync(d_out, 0, (size_t)out_size * sizeof(float), stream);
    (void)hipMemsetAsync(d_ws, 0, cntElems * sizeof(unsigned int), stream);

    // ---- scatter: one wave per point; grid sized exactly (points*32 threads)
    {
        const int block = 256;
        te_scatter_kernel<16, 64><<<(TE_B << 16) * 32 / block, block, 0, stream>>>(
            F[0], Cc[0], outS[0], cntS[0]);
        te_scatter_kernel<15, 128><<<(TE_B << 15) * 32 / block, block, 0, stream>>>(
            F[1], Cc[1], outS[1], cntS[1]);
        te_scatter_kernel<14, 256><<<(TE_B << 14) * 32 / block, block, 0, stream>>>(
            F[2], Cc[2], outS[2], cntS[2]);
    }

    // ---- normalize: one launch per plane (9 total), 4 elems/thread
    {
        const int block = 256;
        for (int p = 0; p < 3; ++p)
            te_normalize_kernel<64><<<(int)(P[0] / 4 / block), block, 0, stream>>>(
                outS[0] + (size_t)p * P[0], cntS[0] + (size_t)p * Q[0]);
        for (int p = 0; p < 3; ++p)
            te_normalize_kernel<128><<<(int)(P[1] / 4 / block), block, 0, stream>>>(
                outS[1] + (size_t)p * P[1], cntS[1] + (size_t)p * Q[1]);
        for (int p = 0; p < 3; ++p)
            te_normalize_kernel<256><<<(int)(P[2] / 4 / block), block, 0, stream>>>(
                outS[2] + (size_t)p * P[2], cntS[2] + (size_t)p * Q[2]);
    }
}